// HaarInverse3D_66924180407100
// MI455X (gfx1250) — compile-verified
//
#include <hip/hip_runtime.h>

typedef float v2f __attribute__((ext_vector_type(2)));
typedef float v8f __attribute__((ext_vector_type(8)));

#define CH_STRIDE   1048576     // 64*128*128   (input channel plane)
#define OUT_DSTRIDE 65536       // 256*256
#define OUT_BSTRIDE 8388608     // 128*256*256

// Weight of channel c in output phase m = {dd,hy,wx} (wx = bit0 so D-VGPR
// pairs (2t,2t+1) are W-adjacent for float2 stores).
// channel bits: c2 -> D sign, c1 -> W sign, c0 -> H sign.
__device__ __forceinline__ float haarW(int m, int c) {
    if (m >= 8) return 0.0f;                       // unused D rows 8..15
    const int e = (((m >> 2) & (c >> 2))           // dd * c2
                 ^ ( m       & (c >> 1))           // wx * c1
                 ^ ((m >> 1) &  c)) & 1;           // hy * c0
    const float s3 = 0.35355339059327378f;         // (1/sqrt(2))^3
    return e ? -s3 : s3;
}

__global__ __launch_bounds__(256)
void haar_inv3d_wmma_kernel(const float* __restrict__ z, float* __restrict__ out)
{
    const int lane = threadIdx.x & 31;
    const int wave = threadIdx.x >> 5;

    // one wave per coarse (b,d,h) row; 4*64*128 = 32768 rows total
    const int rowId = blockIdx.x * 8 + wave;
    const int h = rowId & 127;
    const int d = (rowId >> 7) & 63;
    const int b = rowId >> 13;

    const int n     = lane & 15;    // A row / B column (voxel slot)
    const int khalf = lane >> 4;    // ISA A/B layout: lane half selects K pair
    const int k0    = khalf * 2;

    // A (16x4 f32, 2 VGPRs): constant +/- s^3 weights, K = channel index
    v2f A0, A1;
    A0.x = haarW(n, k0);     A0.y = haarW(n, k0 + 1);   // channels 0..3
    A1.x = haarW(n, k0 + 4); A1.y = haarW(n, k0 + 5);   // channels 4..7

    const float* zin = z + (b * 512 + d) * 16384 + h * 128;       // ch0 plane, w=0
    const int ob0 = b * OUT_BSTRIDE + (2 * d) * OUT_DSTRIDE + (2 * h) * 256;

    for (int chunk = 0; chunk < 8; ++chunk) {
        const int w = chunk * 16 + n;

        // B (4x16 f32, 2 VGPRs per WMMA): row K = channel, column = voxel.
        // lanes 0-15 feed channels {k0, k0+1}, lanes 16-31 {k0, k0+1} with k0=2.
        v2f B0, B1;
        B0.x = __builtin_nontemporal_load(zin + (k0 + 0) * CH_STRIDE + w);
        B0.y = __builtin_nontemporal_load(zin + (k0 + 1) * CH_STRIDE + w);
        B1.x = __builtin_nontemporal_load(zin + (k0 + 4) * CH_STRIDE + w);
        B1.y = __builtin_nontemporal_load(zin + (k0 + 5) * CH_STRIDE + w);

        // D = A0*B0 + A1*B1 : 8 valid rows x 16 voxels, exact fp32
        v8f acc = {};
        acc = __builtin_amdgcn_wmma_f32_16x16x4_f32(
                  false, A0, false, B0, (short)0, acc, false, false);
        acc = __builtin_amdgcn_wmma_f32_16x16x4_f32(
                  false, A1, false, B1, (short)0, acc, false, false);

        // D VGPR r, lanes 0-15 = out phase m=r for voxel n. Pair (2t,2t+1)
        // differs only in wx -> contiguous float2; 16 lanes -> 128B/store.
        if (lane < 16) {
            const int wb = ob0 + 2 * w;
            #pragma unroll
            for (int t = 0; t < 4; ++t) {
                const int dd = t >> 1, hy = t & 1;
                v2f v; v.x = acc[2 * t]; v.y = acc[2 * t + 1];
                __builtin_nontemporal_store(
                    v, (v2f*)(out + wb + dd * OUT_DSTRIDE + hy * 256));
            }
        }
    }
}

extern "C" void kernel_launch(void* const* d_in, const int* in_sizes, int n_in,
                              void* d_out, int out_size, void* d_ws, size_t ws_size,
                              hipStream_t stream) {
    const float* z  = (const float*)d_in[0];   // [4, 8, 64, 128, 128] fp32
    float* out      = (float*)d_out;           // [4, 1, 128, 256, 256] fp32
    (void)in_sizes; (void)n_in; (void)out_size; (void)d_ws; (void)ws_size;

    haar_inv3d_wmma_kernel<<<dim3(4096), dim3(256), 0, stream>>>(z, out);
}